// GCN_HL01_bn_tanh_42545946034236
// MI455X (gfx1250) — compile-verified
//
#include <hip/hip_runtime.h>
#include <math.h>

// GCN: GraphConv(64->128) -> BN -> tanh -> GraphConv(128->32)
// N=50000 nodes, E=800000 edges. Memory-bound by edge scatter; GEMMs on
// fp32 WMMA (v_wmma_f32_16x16x4_f32), exact fp32 math.

#define F1 64
#define F2 128
#define F3 32
#define GCN_BN_EPS 1e-5f

typedef float v2f __attribute__((ext_vector_type(2)));
typedef float v8f __attribute__((ext_vector_type(8)));

// ---------------------------------------------------------------- zero
__global__ void k_zero(float* __restrict__ p, int n) {
  int i = blockIdx.x * blockDim.x + threadIdx.x;
  if (i < n) p[i] = 0.0f;
}

// ---------------------------------------------------------------- scatter-add
// agg[dst] += feat[src] * w[e]; one thread per (edge, 4-feature group).
template <int F>
__global__ void k_scatter(const float* __restrict__ feat,
                          const float* __restrict__ ew,
                          const int* __restrict__ srcv,
                          const int* __restrict__ dstv,
                          float* __restrict__ agg, int E) {
  const int GPE = F / 4;
  int tid = blockIdx.x * blockDim.x + threadIdx.x;
  if (tid >= E * GPE) return;
  int e = tid / GPE;
  int g = (tid % GPE) * 4;
  int s = srcv[e];
  int d = dstv[e];
  float w = ew[e];
  float4 v = *reinterpret_cast<const float4*>(feat + (size_t)s * F + g);
  float* o = agg + (size_t)d * F + g;
  atomicAdd(o + 0, v.x * w);
  atomicAdd(o + 1, v.y * w);
  atomicAdd(o + 2, v.z * w);
  atomicAdd(o + 3, v.w * w);
}

// ---------------------------------------------------------------- WMMA tile MMA
// C(16x16,f32) += A(16xK, row-major lda) * B(Kx16 slice of row-major ldb).
// fp32 WMMA layout (ISA 7.12.2):
//   A 16x4: lane L holds M=L%16,, VGPR0/1 = K = 2*(L/16), 2*(L/16)+1
//   B 4x16 (symmetric): lane L holds N=L%16, K rows 2*(L/16), 2*(L/16)+1
//   C/D  : VGPR r, lane L -> M = r + 8*(L/16), N = L%16
__device__ __forceinline__ v8f wmma_f32_tile(const float* __restrict__ A, int lda,
                                             const float* __restrict__ B, int ldb,
                                             int K, v8f c, int lane) {
  const int m = lane & 15;
  const int n = lane & 15;
  const int koff = (lane >> 4) << 1;
  for (int k = 0; k < K; k += 4) {
    v2f a, b;
    a[0] = A[m * lda + k + koff];
    a[1] = A[m * lda + k + koff + 1];
    b[0] = B[(k + koff) * ldb + n];
    b[1] = B[(k + koff + 1) * ldb + n];
    c = __builtin_amdgcn_wmma_f32_16x16x4_f32(false, a, false, b, (short)0, c,
                                              false, false);
  }
  return c;
}

// ---------------------------------------------------------------- GEMM 1
// h = agg1 @ W1_rel + b1 + x @ W1_root   (N x 64) @ (64 x 128) -> N x 128
// One block per 16-row stripe; 8 waves cover the 8 column tiles; A tiles
// staged once in LDS (stride 68 to dodge bank conflicts, keeps 16B align).
__global__ void k_gemm1(const float* __restrict__ agg,
                        const float* __restrict__ x,
                        const float* __restrict__ Wrel,
                        const float* __restrict__ bias,
                        const float* __restrict__ Wroot,
                        float* __restrict__ h, int N) {
  __shared__ float sAgg[16][F1 + 4];
  __shared__ float sX[16][F1 + 4];
  const int n0 = blockIdx.x * 16;
  {
    int t = threadIdx.x;            // 256 threads == 16 rows x 16 float4
    int row = t >> 4;
    int col = (t & 15) << 2;
    *reinterpret_cast<float4*>(&sAgg[row][col]) =
        *reinterpret_cast<const float4*>(agg + (size_t)(n0 + row) * F1 + col);
    *reinterpret_cast<float4*>(&sX[row][col]) =
        *reinterpret_cast<const float4*>(x + (size_t)(n0 + row) * F1 + col);
  }
  __syncthreads();
  const int lane = threadIdx.x & 31;
  const int j0 = (threadIdx.x >> 5) * 16;   // wave id -> column tile
  v8f c = {};
  c = wmma_f32_tile(&sAgg[0][0], F1 + 4, Wrel + j0, F2, F1, c, lane);
  c = wmma_f32_tile(&sX[0][0],   F1 + 4, Wroot + j0, F2, F1, c, lane);
  const int n = lane & 15;
  const int half = lane >> 4;
  const float bj = bias[j0 + n];
  for (int r = 0; r < 8; ++r) {
    int row = r + half * 8;
    h[(size_t)(n0 + row) * F2 + j0 + n] = c[r] + bj;
  }
}

// ---------------------------------------------------------------- BN stats
// stats[0:128] = sum_f, stats[128:256] = sumsq_f over the N rows of h.
__global__ void k_bn_stats(const float* __restrict__ h,
                           float* __restrict__ stats, int N) {
  __shared__ float s_sum[F2];
  __shared__ float s_sq[F2];
  for (int i = threadIdx.x; i < F2; i += blockDim.x) {
    s_sum[i] = 0.0f;
    s_sq[i] = 0.0f;
  }
  __syncthreads();
  int total = N * F2;
  for (int idx = blockIdx.x * blockDim.x + threadIdx.x; idx < total;
       idx += gridDim.x * blockDim.x) {
    float v = h[idx];
    int f = idx & (F2 - 1);
    atomicAdd(&s_sum[f], v);
    atomicAdd(&s_sq[f], v * v);
  }
  __syncthreads();
  for (int i = threadIdx.x; i < F2; i += blockDim.x) {
    atomicAdd(&stats[i], s_sum[i]);
    atomicAdd(&stats[F2 + i], s_sq[i]);
  }
}

// ---------------------------------------------------------------- BN apply + tanh
__global__ void k_bn_apply(float* __restrict__ h,
                           const float* __restrict__ stats,
                           const float* __restrict__ gamma,
                           const float* __restrict__ beta, int N) {
  int idx = blockIdx.x * blockDim.x + threadIdx.x;
  if (idx >= N * F2) return;
  int f = idx & (F2 - 1);
  float invN = 1.0f / (float)N;
  float mean = stats[f] * invN;
  float var = stats[F2 + f] * invN - mean * mean;
  float v = (h[idx] - mean) * rsqrtf(var + GCN_BN_EPS) * gamma[f] + beta[f];
  h[idx] = tanhf(v);
}

// ---------------------------------------------------------------- GEMM 2
// out = agg2 @ W2_rel + b2 + h @ W2_root   (N x 128) @ (128 x 32) -> N x 32
// One wave per 16x16 output tile (2 column tiles); direct global reads,
// weights (16 KB x2) live in cache.
__global__ void k_gemm2(const float* __restrict__ agg,
                        const float* __restrict__ h,
                        const float* __restrict__ Wrel,
                        const float* __restrict__ bias,
                        const float* __restrict__ Wroot,
                        float* __restrict__ out, int N) {
  int gwave = (blockIdx.x * blockDim.x + threadIdx.x) >> 5;
  const int nTilesJ = F3 / 16;   // 2
  int totalWaves = (N / 16) * nTilesJ;
  if (gwave >= totalWaves) return;       // uniform per wave -> EXEC stays full
  int n0 = (gwave / nTilesJ) * 16;
  int j0 = (gwave % nTilesJ) * 16;
  int lane = threadIdx.x & 31;
  v8f c = {};
  c = wmma_f32_tile(agg + (size_t)n0 * F2, F2, Wrel + j0, F3, F2, c, lane);
  c = wmma_f32_tile(h + (size_t)n0 * F2,   F2, Wroot + j0, F3, F2, c, lane);
  const int n = lane & 15;
  const int half = lane >> 4;
  const float bj = bias[j0 + n];
  for (int r = 0; r < 8; ++r) {
    int row = r + half * 8;
    out[(size_t)(n0 + row) * F3 + j0 + n] = c[r] + bj;
  }
}

// ---------------------------------------------------------------- launch
extern "C" void kernel_launch(void* const* d_in, const int* in_sizes, int n_in,
                              void* d_out, int out_size, void* d_ws, size_t ws_size,
                              hipStream_t stream) {
  const float* x       = (const float*)d_in[0];   // N x 64
  const float* ew      = (const float*)d_in[1];   // E
  const float* W1_rel  = (const float*)d_in[2];   // 64 x 128
  const float* b1      = (const float*)d_in[3];   // 128
  const float* W1_root = (const float*)d_in[4];   // 64 x 128
  const float* gamma   = (const float*)d_in[5];   // 128
  const float* beta    = (const float*)d_in[6];   // 128
  const float* W2_rel  = (const float*)d_in[7];   // 128 x 32
  const float* b2      = (const float*)d_in[8];   // 32
  const float* W2_root = (const float*)d_in[9];   // 128 x 32
  const int*   eidx    = (const int*)d_in[10];    // 2 x E (src row, dst row)

  const int N = in_sizes[0] / F1;
  const int E = in_sizes[1];
  const int* srcv = eidx;
  const int* dstv = eidx + E;

  // workspace layout (floats): agg1 | agg2 | stats(256) | h
  float* agg1  = (float*)d_ws;                    // N*64
  float* agg2  = agg1 + (size_t)N * F1;           // N*128
  float* stats = agg2 + (size_t)N * F2;           // 256
  float* h     = stats + 256;                     // N*128

  const int TPB = 256;

  // 0) zero the atomic accumulators (agg1, agg2, stats are contiguous)
  int nz = N * (F1 + F2) + 256;
  k_zero<<<(nz + TPB - 1) / TPB, TPB, 0, stream>>>(agg1, nz);

  // 1) layer-1 edge scatter: agg1[dst] += x[src] * w
  int t1 = E * (F1 / 4);
  k_scatter<F1><<<(t1 + TPB - 1) / TPB, TPB, 0, stream>>>(x, ew, srcv, dstv, agg1, E);

  // 2) h = agg1 @ W1_rel + b1 + x @ W1_root   (N % 16 == 0 for this problem)
  k_gemm1<<<N / 16, TPB, 0, stream>>>(agg1, x, W1_rel, b1, W1_root, h, N);

  // 3) batch-norm statistics
  k_bn_stats<<<512, TPB, 0, stream>>>(h, stats, N);

  // 4) normalize + affine + tanh (in place)
  int t4 = N * F2;
  k_bn_apply<<<(t4 + TPB - 1) / TPB, TPB, 0, stream>>>(h, stats, gamma, beta, N);

  // 5) layer-2 edge scatter: agg2[dst] += h[src] * w
  int t5 = E * (F2 / 4);
  k_scatter<F2><<<(t5 + TPB - 1) / TPB, TPB, 0, stream>>>(h, ew, srcv, dstv, agg2, E);

  // 6) out = agg2 @ W2_rel + b2 + h @ W2_root
  int waves2 = (N / 16) * (F3 / 16);
  int t6 = waves2 * 32;
  k_gemm2<<<(t6 + TPB - 1) / TPB, TPB, 0, stream>>>(agg2, h, W2_rel, b2, W2_root,
                                                    (float*)d_out, N);
}